// Memory_bank_65180423684843
// MI455X (gfx1250) — compile-verified
//
#include <hip/hip_runtime.h>
#include <hip/hip_bf16.h>

// Roofline: 131072x1024 f32 = 512MB mandatory read -> ~22us at 23.3TB/s.
// Single HBM pass: async-copy 16-row tiles into LDS (ASYNCcnt path),
// WMMA f32 16x16x4 row-sums for the means, VALU for |(x-mu)^2 - my_j|.
// Selection = bitonic keep-32 tree (global top-32 subset of chunk top-32s).

#define DIM        1024
#define NROWS      131072
#define KNN        32
#define TILE_ROWS  16
#define THREADS    256
#define INF_F      3.0e38f

typedef __attribute__((ext_vector_type(2))) float v2f;
typedef __attribute__((ext_vector_type(8))) float v8f;
typedef int v4i __attribute__((__vector_size__(16)));
typedef __attribute__((address_space(1))) v4i glob_v4i;
typedef __attribute__((address_space(3))) v4i lds_v4i;

#if __has_builtin(__builtin_amdgcn_wmma_f32_16x16x4_f32)
#define HAVE_WMMA_F32X4 1
#else
#define HAVE_WMMA_F32X4 0
#endif

#if __has_builtin(__builtin_amdgcn_global_load_async_to_lds_b128)
#define HAVE_ASYNC_LDS 1
#else
#define HAVE_ASYNC_LDS 0
#endif

// LDS tile index with quad-XOR swizzle: element (r,c) lives at
// r*DIM + (((c>>2) ^ r)<<2 | (c&3)).  Keeps WMMA A-operand reads
// (column-strided) and phase-B float4 streams conflict-free.
__device__ __forceinline__ int tswz(int r, int c) {
  return r * DIM + ((((c >> 2) ^ r) << 2) | (c & 3));
}

// ---------------- Kernel 1: query central moments my[j] = (q_j - mu)^2 ----
__global__ void mb_query_moments(const float* __restrict__ q,
                                 float* __restrict__ my) {
  __shared__ float wsum[8];
  __shared__ float mu_s;
  int t = threadIdx.x;                    // 256 threads, 4 floats each
  float4 x = ((const float4*)q)[t];
  float s = x.x + x.y + x.z + x.w;
  for (int off = 16; off > 0; off >>= 1) s += __shfl_xor(s, off, 32);
  if ((t & 31) == 0) wsum[t >> 5] = s;
  __syncthreads();
  if (t == 0) {
    float tot = 0.f;
    for (int i = 0; i < 8; ++i) tot += wsum[i];
    mu_s = tot * (1.0f / DIM);
  }
  __syncthreads();
  float mu = mu_s;
  float4 o;
  o.x = (x.x - mu) * (x.x - mu);
  o.y = (x.y - mu) * (x.y - mu);
  o.z = (x.z - mu) * (x.z - mu);
  o.w = (x.w - mu) * (x.w - mu);
  ((float4*)my)[t] = o;
}

// ---------------- Kernel 2: kl + per-class keying (main, BW-bound) --------
__global__ void __launch_bounds__(THREADS)
mb_kl_kernel(const float* __restrict__ queue,
             const float* __restrict__ my,
             const int* __restrict__ labels,
             float* __restrict__ klA,
             float* __restrict__ klB) {
  __shared__ float tile[TILE_ROWS * DIM];   // 64KB, swizzled
  __shared__ float psum[8 * TILE_ROWS];
  __shared__ float means_s[TILE_ROWS];

  const int t    = threadIdx.x;
  const int wv   = t >> 5;
  const int lane = t & 31;

  // warm the label cacheline for this block's rows (global_prefetch_b8)
  __builtin_prefetch(labels + blockIdx.x * TILE_ROWS, 0, 0);

  // --- stage tile into LDS (swizzled). Prefer CDNA5 async global->LDS ---
  const float* src = queue + (size_t)blockIdx.x * TILE_ROWS * DIM;
#if HAVE_ASYNC_LDS
  for (int i = 0; i < 16; ++i) {
    int idx = i * THREADS + t;            // float4 index, 0..4095
    int r   = idx >> 8;                   // 256 float4 per row
    int c4  = idx & 255;
    __builtin_amdgcn_global_load_async_to_lds_b128(
        (glob_v4i*)(src + (size_t)idx * 4),
        (lds_v4i*)&tile[r * DIM + ((c4 ^ r) << 2)],
        /*offset=*/0, /*cpol=*/0);
  }
#if __has_builtin(__builtin_amdgcn_s_wait_asynccnt)
  __builtin_amdgcn_s_wait_asynccnt(0);
#else
  asm volatile("s_wait_asynccnt 0x0" ::: "memory");
#endif
#else
  for (int i = 0; i < 16; ++i) {
    int idx = i * THREADS + t;            // float4 index, 0..4095
    int r   = idx >> 8;                   // 256 float4 per row
    int c4  = idx & 255;
    float4 v = ((const float4*)src)[idx];
    *(float4*)&tile[r * DIM + ((c4 ^ r) << 2)] = v;
  }
#endif
  __syncthreads();

  // --- phase A: row means.  WMMA f32 16x16x4 with B == ones => C = rowsum.
#if HAVE_WMMA_F32X4
  {
    v2f ones = {1.0f, 1.0f};
    v8f acc  = {0.f, 0.f, 0.f, 0.f, 0.f, 0.f, 0.f, 0.f};
    const int r  = lane & 15;             // A layout: M = lane%16
    const int kb = (lane >> 4) << 1;      // lanes 0-15 -> K {0,1}; 16-31 -> {2,3}
    for (int c = 0; c < 32; ++c) {        // wave wv covers cols [128*wv, 128*wv+128)
      int col = (wv << 7) + (c << 2) + kb;
      int sw  = tswz(r, col);             // col even -> sw even; col+1 is sw|1
      v2f A = {tile[sw], tile[sw | 1]};
      acc = __builtin_amdgcn_wmma_f32_16x16x4_f32(
          /*neg_a=*/false, A, /*neg_b=*/false, ones,
          /*c_mod=*/(short)0, acc, /*reuse_a=*/false, /*reuse_b=*/false);
    }
    // C layout: VGPR m on lanes 0-15 = row m, lanes 16-31 = row m+8.
    if (lane == 0)
      for (int m = 0; m < 8; ++m) psum[wv * TILE_ROWS + m] = acc[m];
    if (lane == 16)
      for (int m = 0; m < 8; ++m) psum[wv * TILE_ROWS + 8 + m] = acc[m];
  }
  __syncthreads();
  if (t < TILE_ROWS) {
    float s = 0.f;
    for (int w2 = 0; w2 < 8; ++w2) s += psum[w2 * TILE_ROWS + t];
    means_s[t] = s * (1.0f / DIM);
  }
  __syncthreads();
#else
  if (t < TILE_ROWS) {
    float s = 0.f;
    for (int c = 0; c < DIM; ++c) s += tile[tswz(t, c)];
    means_s[t] = s * (1.0f / DIM);
  }
  __syncthreads();
#endif

  // --- phase B: kl = sum_j |(x - mu)^2 - my_j|, half-wave per row ---
  {
    const int row = (wv << 1) + (lane >> 4);
    const int li  = lane & 15;
    const float mu = means_s[row];
    float s = 0.f;
    for (int i = 0; i < 16; ++i) {
      int c4  = li + (i << 4);
      float4 x  = *(const float4*)&tile[row * DIM + ((c4 ^ row) << 2)];
      float4 m4 = ((const float4*)my)[c4];
      float d;
      d = x.x - mu; s += fabsf(d * d - m4.x);
      d = x.y - mu; s += fabsf(d * d - m4.y);
      d = x.z - mu; s += fabsf(d * d - m4.z);
      d = x.w - mu; s += fabsf(d * d - m4.w);
    }
    for (int off = 8; off > 0; off >>= 1) s += __shfl_xor(s, off, 32);
    if (li == 0) {
      int grow = blockIdx.x * TILE_ROWS + row;
      int lab  = labels[grow];
      klA[grow] = (lab == 0) ? s : INF_F;
      klB[grow] = (lab == 1) ? s : INF_F;
    }
  }
}

// ---------------- Kernel 3: bitonic sort-1024, keep smallest 32 -----------
__global__ void mb_select32(const float* __restrict__ in,
                            float* __restrict__ out,
                            int inPerClass, int chunksPerClass) {
  __shared__ float sv[1024];
  int cls   = blockIdx.x / chunksPerClass;
  int chunk = blockIdx.x % chunksPerClass;
  const float* base = in + (size_t)cls * inPerClass + (size_t)chunk * 1024;
  int n = inPerClass - chunk * 1024; if (n > 1024) n = 1024;
  int t = threadIdx.x;
  for (int i = t; i < 1024; i += THREADS) sv[i] = (i < n) ? base[i] : INF_F;
  __syncthreads();
  for (int k = 2; k <= 1024; k <<= 1)
    for (int j = k >> 1; j > 0; j >>= 1) {
      for (int i = t; i < 1024; i += THREADS) {
        int ij = i ^ j;
        if (ij > i) {
          float a = sv[i], b = sv[ij];
          bool up = ((i & k) == 0);
          if ((a > b) == up) { sv[i] = b; sv[ij] = a; }
        }
      }
      __syncthreads();
    }
  if (t < KNN)
    out[(size_t)cls * (chunksPerClass * KNN) + chunk * KNN + t] = sv[t];
}

// ---------------- Kernel 4: final 2x128 -> pred + L1-normalized score -----
__global__ void mb_finalize(const float* __restrict__ cand,  // [2][128]
                            float* __restrict__ out) {       // [3]
  __shared__ float sv[256];
  int t = threadIdx.x;                    // 256 threads
  sv[t] = cand[t];
  __syncthreads();
  int cls = t >> 7, i = t & 127;
  for (int k = 2; k <= 128; k <<= 1)
    for (int j = k >> 1; j > 0; j >>= 1) {
      int ij = i ^ j;
      if (ij > i) {
        int ai = cls * 128 + i, bi = cls * 128 + ij;
        float a = sv[ai], b = sv[bi];
        bool up = ((i & k) == 0);
        if ((a > b) == up) { sv[ai] = b; sv[bi] = a; }
      }
      __syncthreads();
    }
  if (t == 0) {
    float dA = 0.f, dB = 0.f;
    for (int q = 0; q < KNN; ++q) { dA += sv[q]; dB += sv[128 + q]; }
    dA *= (1.0f / KNN); dB *= (1.0f / KNN);
    // global 32-NN = merge of the two sorted per-class top-32 lists
    int ia = 0, ib = 0;
    for (int q = 0; q < KNN; ++q) {
      float va = (ia < KNN) ? sv[ia]       : INF_F;
      float vb = (ib < KNN) ? sv[128 + ib] : INF_F;
      if (va <= vb) ++ia; else ++ib;
    }
    int pred = (ib > ia) ? 1 : 0;         // argmax, ties -> class 0
    float denom = fabsf(dA) + fabsf(dB);
    denom = fmaxf(denom, 1e-12f);
    out[0] = (float)pred;
    out[1] = dA / denom;
    out[2] = dB / denom;
  }
}

extern "C" void kernel_launch(void* const* d_in, const int* in_sizes, int n_in,
                              void* d_out, int out_size, void* d_ws, size_t ws_size,
                              hipStream_t stream) {
  const float* query  = (const float*)d_in[0];   // [1,1024] f32
  const float* queue  = (const float*)d_in[1];   // [131072,1024] f32
  const int*   labels = (const int*)d_in[2];     // [131072] int

  float* ws    = (float*)d_ws;
  float* my    = ws;                      // 1024
  float* keyed = my + DIM;                // 2 * 131072 (class0 then class1)
  float* cand1 = keyed + 2 * NROWS;       // 2 * 4096
  float* cand2 = cand1 + 2 * 4096;        // 2 * 128
  float* out   = (float*)d_out;           // [pred, score0, score1]

  mb_query_moments<<<1, THREADS, 0, stream>>>(query, my);
  mb_kl_kernel<<<NROWS / TILE_ROWS, THREADS, 0, stream>>>(
      queue, my, labels, keyed, keyed + NROWS);
  mb_select32<<<2 * (NROWS / 1024), THREADS, 0, stream>>>(
      keyed, cand1, NROWS, NROWS / 1024);                 // 131072 -> 2x4096
  mb_select32<<<2 * 4, THREADS, 0, stream>>>(
      cand1, cand2, 4096, 4);                             // 4096 -> 2x128
  mb_finalize<<<1, THREADS, 0, stream>>>(cand2, out);
}